// BeliefPropagation_61564061221583
// MI455X (gfx1250) — compile-verified
//
#include <hip/hip_runtime.h>

#define D 4096
#define DD ((size_t)D * (size_t)D)

typedef __attribute__((ext_vector_type(2))) float v2f;
typedef __attribute__((ext_vector_type(8))) float v8f;

__global__ void zero_kernel(float* p, int n) {
  int i = blockIdx.x * blockDim.x + threadIdx.x;
  if (i < n) p[i] = 0.f;
}

// Column sums via V_WMMA_F32_16X16X4_F32 with A = ones(16x4):
// D[i,n] = sum_k B[k,n]  -> every row of the 16x16 accumulator holds the
// running column sums. Since A is all-ones, the K->slot mapping of B is
// irrelevant to the result; only the N = lane&15 column mapping matters.
// grid.x = 128 blocks * 8 waves = 1024 waves; wave = {256 col-groups x 4 row-chunks}
// grid.y selects which matrix (theta0 -> c0, theta1 -> c1).
__global__ void colsum_wmma(const float* __restrict__ t0,
                            const float* __restrict__ t1,
                            float* __restrict__ c0,
                            float* __restrict__ c1) {
  const float* in = (blockIdx.y == 0) ? t0 : t1;
  float* out      = (blockIdx.y == 0) ? c0 : c1;
  int wave = blockIdx.x * (blockDim.x >> 5) + (threadIdx.x >> 5);
  int lane = threadIdx.x & 31;
  int colGroup = wave & 255;   // 256 groups of 16 columns
  int rowChunk = wave >> 8;    // 4 chunks of 1024 rows
  int z0 = colGroup << 4;
  int x0 = rowChunk << 10;
  int half = lane >> 4;        // 0 or 1
  int n = lane & 15;           // column within group
  const float* base = in + (size_t)x0 * D + z0 + n;
  v2f a = {1.f, 1.f};          // A-matrix: all ones
  v8f acc = {};                // C accumulator starts at zero
  for (int x = 0; x < 1024; x += 4) {
    const float* p = base + (size_t)(x + 2 * half) * D;
    v2f b;
    b.x = p[0];   // one row of the 4xK chunk (64B segment per half-wave)
    b.y = p[D];   // next row
    acc = __builtin_amdgcn_wmma_f32_16x16x4_f32(
        false, a, false, b, (short)0, acc, false, false);
  }
  // C VGPR0: lanes 0-15 hold row M=0, N=lane -> the column sums.
  if (lane < 16) atomicAdd(&out[z0 + n], acc[0]);
}

// out[y,:] = in[y,:] + rowAdd[y] + colAdd[:]; optionally rowSum[y] = sum(in[y,:]).
// One block (256 threads) per row; float4 coalesced traffic; no atomics.
__global__ void row_pass(const float* __restrict__ in, float* __restrict__ out,
                         const float* __restrict__ rowAdd,
                         const float* __restrict__ colAdd,
                         float* __restrict__ rowSum) {
  int y = blockIdx.x;
  int tid = threadIdx.x;
  float radd = rowAdd ? rowAdd[y] : 0.f;
  const float4* inv = (const float4*)(in + (size_t)y * D);
  float4* outv = (float4*)(out + (size_t)y * D);
  const float4* cav = (const float4*)colAdd;
  float acc = 0.f;
  for (int i = tid; i < D / 4; i += 256) {
    float4 v = inv[i];
    acc += (v.x + v.y) + (v.z + v.w);
    float4 r;
    if (cav) {
      float4 c = cav[i];
      r.x = v.x + radd + c.x; r.y = v.y + radd + c.y;
      r.z = v.z + radd + c.z; r.w = v.w + radd + c.w;
    } else {
      r.x = v.x + radd; r.y = v.y + radd; r.z = v.z + radd; r.w = v.w + radd;
    }
    outv[i] = r;
  }
  if (rowSum) {
    for (int off = 16; off > 0; off >>= 1) acc += __shfl_xor(acc, off, 32);
    __shared__ float red[8];
    int w = tid >> 5, l = tid & 31;
    if (l == 0) red[w] = acc;
    __syncthreads();
    if (tid < 8) {
      float r = red[tid];
      for (int off = 4; off > 0; off >>= 1) r += __shfl_xor(r, off, 32);
      if (tid == 0) rowSum[y] = r;
    }
  }
}

// m0 = scatter(c0, op0); S0 = sum(m0); m1 = scatter(c1 + S0, op1)
__global__ void vec_phase1(const float* __restrict__ c0, const float* __restrict__ c1,
                           const int* __restrict__ op0, const int* __restrict__ op1,
                           float* __restrict__ m0, float* __restrict__ m1) {
  __shared__ float sm[D];
  __shared__ float red[32];
  __shared__ float sb;
  int tid = threadIdx.x;  // 1024 threads
  for (int i = tid; i < D; i += 1024) sm[i] = 0.f;
  __syncthreads();
  for (int i = tid; i < D; i += 1024) atomicAdd(&sm[op0[D + i]], c0[op0[i]]);
  __syncthreads();
  float part = 0.f;
  for (int i = tid; i < D; i += 1024) { float v = sm[i]; part += v; m0[i] = v; }
  for (int off = 16; off > 0; off >>= 1) part += __shfl_xor(part, off, 32);
  if ((tid & 31) == 0) red[tid >> 5] = part;
  __syncthreads();
  if (tid < 32) {
    float r = red[tid];
    for (int off = 16; off > 0; off >>= 1) r += __shfl_xor(r, off, 32);
    if (tid == 0) sb = r;
  }
  __syncthreads();
  float S0 = sb;
  for (int i = tid; i < D; i += 1024) sm[i] = 0.f;
  __syncthreads();
  for (int i = tid; i < D; i += 1024) atomicAdd(&sm[op1[D + i]], c1[op1[i]] + S0);
  __syncthreads();
  for (int i = tid; i < D; i += 1024) m1[i] = sm[i];
}

// m2 = scatter(r2 + 4095*m1, op2); S2 = sum(m2)
__global__ void vec_phase2(const float* __restrict__ r2, const float* __restrict__ m1,
                           const int* __restrict__ op2,
                           float* __restrict__ m2, float* __restrict__ S2out) {
  __shared__ float sm[D];
  __shared__ float red[32];
  int tid = threadIdx.x;
  for (int i = tid; i < D; i += 1024) sm[i] = 0.f;
  __syncthreads();
  for (int i = tid; i < D; i += 1024) {
    int j = op2[i];
    atomicAdd(&sm[op2[D + i]], r2[j] + 4095.0f * m1[j]);
  }
  __syncthreads();
  float part = 0.f;
  for (int i = tid; i < D; i += 1024) { float v = sm[i]; part += v; m2[i] = v; }
  for (int off = 16; off > 0; off >>= 1) part += __shfl_xor(part, off, 32);
  if ((tid & 31) == 0) red[tid >> 5] = part;
  __syncthreads();
  if (tid < 32) {
    float r = red[tid];
    for (int off = 16; off > 0; off >>= 1) r += __shfl_xor(r, off, 32);
    if (tid == 0) S2out[0] = r;
  }
}

// m3 = scatter(r1 + 4095*m0 + S2, op3)
__global__ void vec_phase3(const float* __restrict__ r1, const float* __restrict__ m0,
                           const float* __restrict__ S2in, const int* __restrict__ op3,
                           float* __restrict__ m3) {
  __shared__ float sm[D];
  int tid = threadIdx.x;
  float S2 = S2in[0];
  for (int i = tid; i < D; i += 1024) sm[i] = 0.f;
  __syncthreads();
  for (int i = tid; i < D; i += 1024) {
    int j = op3[i];
    atomicAdd(&sm[op3[D + i]], r1[j] + 4095.0f * m0[j] + S2);
  }
  __syncthreads();
  for (int i = tid; i < D; i += 1024) m3[i] = sm[i];
}

extern "C" void kernel_launch(void* const* d_in, const int* in_sizes, int n_in,
                              void* d_out, int out_size, void* d_ws, size_t ws_size,
                              hipStream_t stream) {
  const float* theta0 = (const float*)d_in[0];
  const float* theta1 = (const float*)d_in[1];
  const float* theta2 = (const float*)d_in[2];
  const int* op0 = (const int*)d_in[3];
  const int* op1 = (const int*)d_in[4];
  const int* op2 = (const int*)d_in[5];
  const int* op3 = (const int*)d_in[6];

  float* out0 = (float*)d_out;       // theta0_out
  float* out1 = out0 + DD;           // theta1_out
  float* out2 = out1 + DD;           // theta2_out

  float* ws = (float*)d_ws;
  float* c0 = ws;                    // colsum(theta0)
  float* c1 = ws + 1 * D;            // colsum(theta1)
  float* r1 = ws + 2 * D;            // rowsum(theta1)
  float* r2 = ws + 3 * D;            // rowsum(theta2)
  float* m0 = ws + 4 * D;
  float* m1 = ws + 5 * D;
  float* m2 = ws + 6 * D;
  float* m3 = ws + 7 * D;
  float* S2 = ws + 8 * D;

  // c0,c1 are atomic targets -> zero them every call (ws is not re-poisoned).
  zero_kernel<<<(2 * D + 255) / 256, 256, 0, stream>>>(c0, 2 * D);

  // WMMA column-sum reductions of theta0 and theta1 (read 128 MB).
  colsum_wmma<<<dim3(128, 2), 256, 0, stream>>>(theta0, theta1, c0, c1);

  // m0, S0, m1 (tiny, single block for intra-stage ordering).
  vec_phase1<<<1, 1024, 0, stream>>>(c0, c1, op0, op1, m0, m1);

  // theta2_out = theta2 + m1[:,None], fused rowsum(theta2) -> r2.
  row_pass<<<D, 256, 0, stream>>>(theta2, out2, m1, nullptr, r2);

  // m2, S2.
  vec_phase2<<<1, 1024, 0, stream>>>(r2, m1, op2, m2, S2);

  // theta1_out = theta1 + m0[:,None] + m2[None,:], fused rowsum(theta1) -> r1.
  row_pass<<<D, 256, 0, stream>>>(theta1, out1, m0, m2, r1);

  // m3.
  vec_phase3<<<1, 1024, 0, stream>>>(r1, m0, S2, op3, m3);

  // theta0_out = theta0 + m3[None,:].
  row_pass<<<D, 256, 0, stream>>>(theta0, out0, nullptr, m3, nullptr);
}